// EpisodicMemory_57810259804539
// MI455X (gfx1250) — compile-verified
//
#include <hip/hip_runtime.h>

typedef float v2f __attribute__((ext_vector_type(2)));
typedef float v8f __attribute__((ext_vector_type(8)));

#define B_   4
#define H_   16
#define S_   2048
#define D_   64
#define M_   1000
#define HID_ 1024
#define K_   10
#define SK_  2058      // S_ + K_
#define EPS_ 1e-8f

// ---------------- bulk float4 copy ----------------
__global__ void copy4_kernel(float4* __restrict__ dst, const float4* __restrict__ src, long n4) {
    long i = (long)blockIdx.x * blockDim.x + threadIdx.x;
    long stride = (long)gridDim.x * blockDim.x;
    for (; i < n4; i += stride) dst[i] = src[i];
}

// k/v -> k_aug/v_aug body (rows K_..K_+S_-1 per (b,h))
__global__ void kv_copy_kernel(float4* __restrict__ kdst, const float4* __restrict__ ksrc,
                               float4* __restrict__ vdst, const float4* __restrict__ vsrc) {
    const long n4 = (long)B_ * H_ * S_ * D_ / 4;   // 2,097,152
    const int per_bh_src = S_ * D_ / 4;            // 32768
    const int per_bh_dst = SK_ * D_ / 4;           // 32928
    const int head4      = K_ * D_ / 4;            // 160
    long i = (long)blockIdx.x * blockDim.x + threadIdx.x;
    long stride = (long)gridDim.x * blockDim.x;
    for (; i < n4; i += stride) {
        int bh   = (int)(i / per_bh_src);
        int rest = (int)(i % per_bh_src);
        long d = (long)bh * per_bh_dst + head4 + rest;
        kdst[d] = ksrc[i];
        vdst[d] = vsrc[i];
    }
}

// mask body + positions_q
__global__ void mask_pos_kernel(const float* __restrict__ mask_in,
                                float* __restrict__ mask_out, float* __restrict__ pos_out) {
    int tid = blockIdx.x * blockDim.x + threadIdx.x;   // 8192 total
    if (tid >= B_ * S_) return;
    int b = tid >> 11;          // / 2048
    int s = tid & (S_ - 1);
    mask_out[b * SK_ + K_ + s] = mask_in[b * S_ + s];
    pos_out[b * SK_ + s] = (float)s;
}

// query_key buffer [16][1024], rows 0..3 = k[b,:, S-1, :] flattened, rows 4..15 zero
__global__ void prep_qk_kernel(const float* __restrict__ kin, float* __restrict__ qk) {
    int tid = blockIdx.x * blockDim.x + threadIdx.x;   // 16384 total
    if (tid >= 16 * HID_) return;
    int r = tid >> 10;
    int c = tid & (HID_ - 1);
    float val = 0.0f;
    if (r < B_) {
        int h = c >> 6;
        int d = c & (D_ - 1);
        val = kin[(((long)(r * H_ + h)) * S_ + (S_ - 1)) * D_ + d];
    }
    qk[tid] = val;
}

// per-row 1/(||row||+eps), rows of length HID_
__global__ void row_norm_kernel(const float* __restrict__ rows, float* __restrict__ inv_norm) {
    __shared__ float sdata[256];
    int row = blockIdx.x;
    int t = threadIdx.x;
    const float* p = rows + (long)row * HID_;
    float s = 0.0f;
    for (int c = t; c < HID_; c += 256) { float v = p[c]; s += v * v; }
    sdata[t] = s;
    __syncthreads();
    for (int off = 128; off > 0; off >>= 1) {
        if (t < off) sdata[t] += sdata[t + off];
        __syncthreads();
    }
    if (t == 0) inv_norm[row] = 1.0f / (sqrtf(sdata[0]) + EPS_);
}

// sims[b][m] via V_WMMA_F32_16X16X4_F32 : one wave per 16 memory rows
__global__ void sims_wmma_kernel(const float* __restrict__ qk,
                                 const float* __restrict__ memk,
                                 const float* __restrict__ invq,
                                 const float* __restrict__ invm,
                                 float* __restrict__ sims) {
    int lane  = threadIdx.x;            // 0..31, wave32
    int mrow  = lane & 15;              // A: M index
    int khalf = (lane >> 4) & 1;        // 0 -> K {k,k+1}, 1 -> K {k+2,k+3}
    int ncol  = blockIdx.x * 16 + (lane & 15);  // B/D: N index (memory row)
    bool nvalid = (ncol < M_);
    const float* arow = qk + (long)mrow * HID_;
    const float* brow = memk + (long)ncol * HID_;
    v8f c = (v8f)0.0f;
    #pragma unroll 4
    for (int kb = 0; kb < HID_; kb += 4) {
        int kk = kb + khalf * 2;
        v2f a; a.x = arow[kk]; a.y = arow[kk + 1];
        v2f b;
        if (nvalid) { b.x = brow[kk]; b.y = brow[kk + 1]; }
        else        { b.x = 0.0f;     b.y = 0.0f; }
        c = __builtin_amdgcn_wmma_f32_16x16x4_f32(false, a, false, b, (short)0, c, false, false);
    }
    // D layout: lanes 0..15, VGPR r -> element (M=r, N=lane); batches are M=0..3
    if (lane < 16 && nvalid) {
        float sm = invm[ncol];
        #pragma unroll
        for (int r = 0; r < B_; ++r)
            sims[r * HID_ + ncol] = c[r] * invq[r] * sm;
    }
}

// stable top-10 per batch (ties -> lower index first, matching jax.lax.top_k)
__global__ void topk_kernel(const float* __restrict__ sims, int* __restrict__ topidx) {
    int b = threadIdx.x;
    if (b >= B_) return;
    float vals[K_]; int ids[K_];
    #pragma unroll
    for (int j = 0; j < K_; ++j) { vals[j] = -3.4e38f; ids[j] = 0; }
    for (int m = 0; m < M_; ++m) {
        float v = sims[b * HID_ + m];
        if (v > vals[K_ - 1]) {
            int j = K_ - 1;
            while (j > 0 && v > vals[j - 1]) { vals[j] = vals[j - 1]; ids[j] = ids[j - 1]; --j; }
            vals[j] = v; ids[j] = m;
        }
    }
    #pragma unroll
    for (int j = 0; j < K_; ++j) topidx[b * K_ + j] = ids[j];
}

// scatter retrieved k/v rows, retrieved positions, mask head, seq_len_k
__global__ void fill_retrieved_kernel(const int* __restrict__ topidx,
                                      const float* __restrict__ memk,
                                      const float* __restrict__ memv,
                                      const float* __restrict__ mempos,
                                      float* __restrict__ kaug, float* __restrict__ vaug,
                                      float* __restrict__ mask, float* __restrict__ pos,
                                      float* __restrict__ slk) {
    int tid = blockIdx.x * blockDim.x + threadIdx.x;
    const int total = B_ * H_ * K_ * D_;     // 40960
    if (tid < total) {
        int d  = tid & (D_ - 1);
        int kk = (tid >> 6) % K_;
        int h  = (tid / (D_ * K_)) % H_;
        int b  = tid / (D_ * K_ * H_);
        int idx = topidx[b * K_ + kk];
        long src = (long)idx * HID_ + h * D_ + d;
        long dst = (((long)(b * H_ + h)) * SK_ + kk) * D_ + d;
        kaug[dst] = memk[src];
        vaug[dst] = memv[src];
    }
    if (tid < B_ * K_) {
        int b = tid / K_, kk = tid % K_;
        int idx = topidx[b * K_ + kk];
        pos[b * SK_ + S_ + kk] = mempos[idx];
        mask[b * SK_ + kk] = 1.0f;
    }
    if (tid == 0) slk[0] = (float)SK_;
}

extern "C" void kernel_launch(void* const* d_in, const int* in_sizes, int n_in,
                              void* d_out, int out_size, void* d_ws, size_t ws_size,
                              hipStream_t stream) {
    const float* in_inputs = (const float*)d_in[0];
    const float* in_q      = (const float*)d_in[1];
    const float* in_k      = (const float*)d_in[2];
    const float* in_v      = (const float*)d_in[3];
    const float* in_mask   = (const float*)d_in[4];
    const float* in_memk   = (const float*)d_in[5];
    const float* in_memv   = (const float*)d_in[6];
    const float* in_mempos = (const float*)d_in[7];
    // d_in[8] = seq_len_q (scalar), value is known statically

    float* out = (float*)d_out;
    const long N_in   = (long)B_ * S_ * H_ * D_;    // 8,388,608
    const long N_kaug = (long)B_ * H_ * SK_ * D_;   // 8,429,568
    float* o_inputs = out;
    float* o_q      = out + N_in;
    float* o_kaug   = out + 2 * N_in;
    float* o_vaug   = o_kaug + N_kaug;
    float* o_mask   = o_vaug + N_kaug;
    float* o_slk    = o_mask + (long)B_ * SK_;
    float* o_pos    = o_slk + 1;

    char* ws = (char*)d_ws;
    float* qk     = (float*)(ws);            // 16*1024 f32 = 64 KB
    float* invm   = (float*)(ws + 65536);    // 1024 f32
    float* invq   = (float*)(ws + 69632);    // 16 f32
    float* sims   = (float*)(ws + 69696);    // 4*1024 f32
    int*   topidx = (int*)  (ws + 86080);    // 40 i32

    // bulk copies (memory-bound part: ~270 MB total traffic)
    copy4_kernel<<<2048, 256, 0, stream>>>((float4*)o_inputs, (const float4*)in_inputs, N_in / 4);
    copy4_kernel<<<2048, 256, 0, stream>>>((float4*)o_q, (const float4*)in_q, N_in / 4);
    kv_copy_kernel<<<2048, 256, 0, stream>>>((float4*)o_kaug, (const float4*)in_k,
                                             (float4*)o_vaug, (const float4*)in_v);
    mask_pos_kernel<<<32, 256, 0, stream>>>(in_mask, o_mask, o_pos);

    // retrieval math (tiny): qk buffer -> norms -> WMMA sims -> top-k -> scatter
    prep_qk_kernel<<<64, 256, 0, stream>>>(in_k, qk);
    row_norm_kernel<<<M_, 256, 0, stream>>>(in_memk, invm);
    row_norm_kernel<<<B_, 256, 0, stream>>>(qk, invq);
    sims_wmma_kernel<<<(M_ + 15) / 16, 32, 0, stream>>>(qk, in_memk, invq, invm, sims);
    topk_kernel<<<1, 32, 0, stream>>>(sims, topidx);
    fill_retrieved_kernel<<<160, 256, 0, stream>>>(topidx, in_memk, in_memv, in_mempos,
                                                   o_kaug, o_vaug, o_mask, o_pos, o_slk);
}